// QTGeometricProduct_266287972627
// MI455X (gfx1250) — compile-verified
//
#include <hip/hip_runtime.h>
#include <math.h>

#define DIMB 64      // blades
#define FD   128     // feature dim
#define BD   256     // batch
#define EPSV 1e-6f
#define NSLOT 80     // class-padded slots per batch (16/16/16/32)

typedef float v2f __attribute__((ext_vector_type(2)));
typedef float v8f __attribute__((ext_vector_type(8)));
typedef unsigned int u32x4 __attribute__((ext_vector_type(4)));
typedef int i32x4 __attribute__((ext_vector_type(4)));
typedef int i32x8 __attribute__((ext_vector_type(8)));

// ---- workspace layout (bytes), one contiguous block DMA'd to LDS ----
// [0,     32768) : w4   f32[128][64]   dense class-tensor per feature
// [32768, 40960) : tbl  u16[j][i] = slot(k) | sign<<7 | clsidx<<8
// [40960, 41280) : s2b  i32[80]  slot  -> blade (-1 = pad)
// [41280, 41536) : b2s  i32[64]  blade -> slot
// [41536, 41792) : clsv i32[64]  blade -> class
#define WS_TBL_BYTES 41792
#define WS_TBL_ELTS  (WS_TBL_BYTES / 4)   // 10448 dwords

// LDS layout inside qt_fused
#define L_XR   0            // 4*80*128 f32 = 163840 B (slot space)
#define L_TBL  163840       // table block (WS_TBL_BYTES)
#define L_W4   (L_TBL + 0)
#define L_T16  (L_TBL + 32768)
#define L_S2B  (L_TBL + 40960)
#define L_B2S  (L_TBL + 41280)
#define L_CLS  (L_TBL + 41536)
#define L_TOTAL (L_TBL + WS_TBL_BYTES)    // 205632 B  (<320KB/WGP)

__global__ void build_tables(const float* __restrict__ cayley,
                             const int* __restrict__ cls,
                             unsigned short* __restrict__ tbl,
                             int* __restrict__ s2b,
                             int* __restrict__ b2s,
                             int* __restrict__ clsv) {
  const int base[4] = {0, 16, 32, 48};
  int tid = blockIdx.x * blockDim.x + threadIdx.x;
  if (tid < DIMB * DIMB) {
    int i = tid & 63, j = tid >> 6;
    int kk = 0; float sg = 1.f;
    for (int k = 0; k < DIMB; ++k) {
      float v = cayley[(size_t)(i * DIMB + j) * DIMB + k];
      if (v != 0.f) { kk = k; sg = v; }
    }
    int ck = cls[kk];
    int rank = 0;
    for (int m = 0; m < DIMB; ++m) rank += (m < kk && cls[m] == ck) ? 1 : 0;
    int slot = base[ck] + rank;                       // 0..79, fits 7 bits
    int cidx = cls[i] * 16 + cls[j] * 4 + ck;         // 0..63
    tbl[j * DIMB + i] =
        (unsigned short)(slot | (sg < 0.f ? 0x80 : 0) | (cidx << 8));
  }
  if (tid < NSLOT) {
    int g = tid >> 4;
    int c = g > 3 ? 3 : g;
    int r = (tid & 15) + (g == 4 ? 16 : 0);
    int blade = -1, cnt = 0;
    for (int i2 = 0; i2 < DIMB; ++i2)
      if (cls[i2] == c) { if (cnt == r) blade = i2; ++cnt; }
    s2b[tid] = blade;
  }
  if (tid < DIMB) {
    int c = cls[tid];
    int rank = 0;
    for (int m = 0; m < tid; ++m) rank += (cls[m] == c) ? 1 : 0;
    b2s[tid]  = base[c] + rank;
    clsv[tid] = c;
  }
}

__global__ void build_w4(const float* __restrict__ w_gp,
                         const int* __restrict__ pidx, int P,
                         float* __restrict__ w4) {
  int n = blockIdx.x;
  int t = threadIdx.x;
  w4[n * DIMB + t] = 0.f;
  __syncthreads();
  for (int p = t; p < P; p += 64) {
    int a = pidx[3 * p], b = pidx[3 * p + 1], c = pidx[3 * p + 2];
    w4[n * DIMB + a * 16 + b * 4 + c] = w_gp[n * P + p];
  }
}

// Fused: per-class linear (WMMA f32 16x16x4) -> gated norm -> geometric product.
// One workgroup = 4 batches; xr tile kept in LDS in padded slot space.
__launch_bounds__(512)
__global__ void qt_fused(const float* __restrict__ x,
                         const float* __restrict__ w_lin,
                         const float* __restrict__ b_lin,
                         const float* __restrict__ a_norm,
                         const void* __restrict__ ws_base,
                         float* __restrict__ out) {
  extern __shared__ char smem[];
  float* xr                 = (float*)(smem + L_XR);
  float* w4                 = (float*)(smem + L_W4);
  const unsigned short* t16 = (const unsigned short*)(smem + L_T16);
  const int* s2b            = (const int*)(smem + L_S2B);
  const int* b2s            = (const int*)(smem + L_B2S);
  const int* clsL           = (const int*)(smem + L_CLS);

  const int t  = threadIdx.x;
  const int b0 = blockIdx.x * 4;

  // ---- table block: global -> LDS via Tensor Data Mover (one descriptor) ----
#if __has_builtin(__builtin_amdgcn_tensor_load_to_lds)
  if (t < 32) {   // one wave issues the DMA and waits on TENSORcnt
    unsigned long long ga = (unsigned long long)ws_base;
    u32x4 g0;
    g0[0] = 1u;                                    // count=1, user descriptor
    g0[1] = (unsigned)L_TBL;                       // lds_addr (bytes)
    g0[2] = (unsigned)(ga & 0xffffffffu);          // global_addr lo
    g0[3] = (unsigned)((ga >> 32) & 0x1ffffffu) | (2u << 30); // hi | type=2
    i32x8 g1;
    g1[0] = 0x00020000;                            // data_size = 4B
    g1[1] = (int)(((unsigned)WS_TBL_ELTS & 0xffffu) << 16);   // tensor_dim0 lo
    g1[2] = (int)(((unsigned)WS_TBL_ELTS >> 16) | (1u << 16)); // dim0 hi|dim1=1
    g1[3] = (int)((unsigned)WS_TBL_ELTS << 16);    // tile_dim0
    g1[4] = 1;                                     // tile_dim1 = 1
    g1[5] = WS_TBL_ELTS;                           // tensor_dim0_stride
    g1[6] = 0; g1[7] = 0;
    i32x4 z4 = {0, 0, 0, 0};
#if __clang_major__ >= 23
    i32x8 z8 = {0, 0, 0, 0, 0, 0, 0, 0};
    __builtin_amdgcn_tensor_load_to_lds(g0, g1, z4, z4, z8, 0);
#else
    __builtin_amdgcn_tensor_load_to_lds(g0, g1, z4, z4, 0);
#endif
    __builtin_amdgcn_s_wait_tensorcnt(0);
  }
#else
  {
    const unsigned* src = (const unsigned*)ws_base;
    unsigned* dst = (unsigned*)(smem + L_TBL);
    for (int q = t; q < WS_TBL_ELTS; q += 512) dst[q] = src[q];
  }
#endif
  __syncthreads();

  // ---------- Stage 1: xr = x @ W_cls^T via V_WMMA_F32_16X16X4_F32 ----------
  const int wave  = t >> 5;
  const int lane  = t & 31;
  const int lrow  = lane & 15;
  const int khalf = (lane >> 4) << 1;      // A/B K offset 0 or 2

  for (int tile = wave; tile < 160; tile += 16) {  // 20 M-tiles x 8 N-tiles
    int mt = tile >> 3;
    int nt = tile & 7;
    int bl = mt / 5;                  // batch within workgroup
    int st = mt % 5;                  // slot group
    int cc = st > 3 ? 3 : st;         // uniform grade-class for this tile
    int blade  = s2b[st * 16 + lrow];
    int sblade = blade < 0 ? 0 : blade;

    const float* xrow = x + ((size_t)(b0 + bl) * FD) * DIMB + sblade;     // +m*64
    const float* wrow = w_lin + ((size_t)(nt * 16 + lrow) * FD) * 4 + cc; // +m*4
    __builtin_prefetch(xrow, 0, 1);   // global_prefetch_b8

    v8f acc = {};
    for (int kk = 0; kk < 32; ++kk) { // K = 128, 4 per WMMA
      int m0 = kk * 4 + khalf;
      v2f a, b;
      a.x = xrow[(size_t)m0 * DIMB];
      a.y = xrow[(size_t)(m0 + 1) * DIMB];
      b.x = wrow[(size_t)m0 * 4];
      b.y = wrow[(size_t)(m0 + 1) * 4];
      acc = __builtin_amdgcn_wmma_f32_16x16x4_f32(false, a, false, b,
                                                  (short)0, acc, false, false);
    }
    int nbase = nt * 16 + lrow;
    int rhalf = (lane >> 4) << 3;     // C rows: lanes 0-15 -> M=v, 16-31 -> v+8
#pragma unroll
    for (int v = 0; v < 8; ++v)       // unconditional: slot-space, pads included
      xr[(bl * NSLOT + st * 16 + v + rhalf) * FD + nbase] = acc[v];
  }
  __syncthreads();

  // ---------- Stage 2+3: one thread per (b_local, n) ----------
  const int bl = t >> 7;
  const int n  = t & 127;
  float* xrc = xr + (bl * NSLOT) * FD + n;    // + slot*128

  xrc[b2s[0] * FD] += b_lin[n];               // bias on scalar blade

  float s0 = 0.f, s1 = 0.f, s2 = 0.f, s3 = 0.f;
  for (int i = 0; i < DIMB; ++i) {
    float v = xrc[b2s[i] * FD];
    float v2 = v * v;
    int c = clsL[i];
    s0 += (c == 0) ? v2 : 0.f;
    s1 += (c == 1) ? v2 : 0.f;
    s2 += (c == 2) ? v2 : 0.f;
    s3 += (c == 3) ? v2 : 0.f;
  }
  float inv[4];
#pragma unroll
  for (int c = 0; c < 4; ++c) {
    float sc = (c == 0) ? s0 : (c == 1) ? s1 : (c == 2) ? s2 : s3;
    float sig = 1.f / (1.f + __expf(-a_norm[n * 4 + c]));
    inv[c] = 1.f / (sig * (sqrtf(sc) - 1.f) + 1.f + EPSV);
  }
  for (int i = 0; i < DIMB; ++i) {
    int c = clsL[i];
    float g = (c == 0) ? inv[0] : (c == 1) ? inv[1] : (c == 2) ? inv[2] : inv[3];
    xrc[b2s[i] * FD] *= g;
  }

  // geometric product: x[b,n,:] in VGPRs, xr gathered from LDS via u16 table
  const int bg = b0 + bl;
  const float* xp = x + ((size_t)bg * FD + n) * DIMB;
  float xv[DIMB];
#pragma unroll
  for (int q = 0; q < 16; ++q) {
    float4 v4 = ((const float4*)xp)[q];
    xv[q * 4 + 0] = v4.x; xv[q * 4 + 1] = v4.y;
    xv[q * 4 + 2] = v4.z; xv[q * 4 + 3] = v4.w;
  }
  const float* w4n = w4 + n * DIMB;
  float* op = out + ((size_t)bg * FD + n) * DIMB;
  for (int j = 0; j < DIMB; ++j) {
    const unsigned short* kt = t16 + j * DIMB;
    float acc = 0.f;
#pragma unroll
    for (int i = 0; i < DIMB; ++i) {
      unsigned e = kt[i];                       // one ds_load_u16 per term
      float prod = w4n[e >> 8] * xv[i] * xrc[(e & 127) * FD];
      acc += (e & 0x80) ? -prod : prod;
    }
    op[j] = acc;
  }
}

extern "C" void kernel_launch(void* const* d_in, const int* in_sizes, int n_in,
                              void* d_out, int out_size, void* d_ws, size_t ws_size,
                              hipStream_t stream) {
  const float* x      = (const float*)d_in[0];
  const float* w_gp   = (const float*)d_in[1];
  const float* w_lin  = (const float*)d_in[2];
  const float* b_lin  = (const float*)d_in[3];
  const float* a_norm = (const float*)d_in[4];
  const float* cayley = (const float*)d_in[5];
  const int*   cls    = (const int*)d_in[6];
  const int*   pidx   = (const int*)d_in[7];
  const int P = in_sizes[7] / 3;

  char* ws = (char*)d_ws;
  float*          w4   = (float*)(ws + 0);
  unsigned short* tbl  = (unsigned short*)(ws + 32768);
  int*            s2b  = (int*)(ws + 40960);
  int*            b2s  = (int*)(ws + 41280);
  int*            clsv = (int*)(ws + 41536);

  build_tables<<<16, 256, 0, stream>>>(cayley, cls, tbl, s2b, b2s, clsv);
  build_w4<<<FD, 64, 0, stream>>>(w_gp, pidx, P, w4);
  qt_fused<<<BD / 4, 512, L_TOTAL, stream>>>(x, w_lin, b_lin, a_norm,
                                             (const void*)ws, (float*)d_out);
}